// Least_Squares_65274912965338
// MI455X (gfx1250) — compile-verified
//
#include <hip/hip_runtime.h>

// Batched 2D Kabsch (B=8192, N=2048): closed-form 2x2 SVD => streaming
// 8-accumulator reduction. Bandwidth-bound: 256 MiB @ 23.3 TB/s ~= 11.5 us.
// CDNA5 paths: global_load_async_to_lds_b128 (ASYNCcnt) for staging, and
// V_WMMA_F32_16X16X4_F32 (B = ones) for the exact f32 cross-lane reduction.

#define NPTS  2048
#define TPB   256
#define F4_PER_BATCH (NPTS / 2)          // 1024 float4 per array per batch
#define CHUNKS (F4_PER_BATCH / TPB)      // 4 float4 chunks per thread per array

typedef __attribute__((ext_vector_type(2))) float v2f;
typedef __attribute__((ext_vector_type(8))) float v8f;

// CDNA5 async copy: global memory -> LDS, tracked with ASYNCcnt.
__device__ __forceinline__ void async_load_f4(unsigned int lds_byte_off,
                                              unsigned long long gaddr) {
  asm volatile("global_load_async_to_lds_b128 %0, %1, off"
               :: "v"(lds_byte_off), "v"(gaddr)
               : "memory");
}

__device__ __forceinline__ void wait_async_zero() {
#if __has_builtin(__builtin_amdgcn_s_wait_asynccnt)
  __builtin_amdgcn_s_wait_asynccnt(0);
#else
  asm volatile("s_wait_asynccnt 0" ::: "memory");
#endif
}

// Exact f32 full-wave (32 lane) sum via V_WMMA_F32_16X16X4_F32.
// A carries the data (documented 16x4 layout: row m in lanes m / m+16,
// VGPR0 = K{0,2}, VGPR1 = K{1,3}); we set VGPR1 = 0 so only K=0,2 contribute.
// B = all ones (layout-invariant). D[m][n] = x(m) + x(m+16) for all n.
// Sum of a lane's 8 D VGPRs = half-row total; one xor-16 shuffle finishes.
__device__ __forceinline__ float wave_reduce_wmma(float x) {
  v2f a;  a[0] = x;     a[1] = 0.0f;
  v2f b1; b1[0] = 1.0f; b1[1] = 1.0f;
  v8f c = {};
  v8f d = __builtin_amdgcn_wmma_f32_16x16x4_f32(
      /*neg_a=*/false, a, /*neg_b=*/false, b1,
      /*c_mod=*/(short)0, c, /*reuse_a=*/false, /*reuse_b=*/false);
  float h = ((d[0] + d[1]) + (d[2] + d[3])) + ((d[4] + d[5]) + (d[6] + d[7]));
  h += __shfl_xor(h, 16, 32);   // add other half's rows -> full 32-lane sum
  return h;
}

__global__ __launch_bounds__(TPB) void kabsch2d_kernel(
    const float* __restrict__ pA, const float* __restrict__ pB,
    float* __restrict__ out, int nbatch) {
  __shared__ float4 sA[F4_PER_BATCH];    // 16 KB
  __shared__ float4 sB[F4_PER_BATCH];    // 16 KB
  __shared__ float  red[8][8];           // [wave][accumulator]

  const int b = blockIdx.x;
  const int t = threadIdx.x;

  const float4* gA = reinterpret_cast<const float4*>(pA) + (size_t)b * F4_PER_BATCH;
  const float4* gB = reinterpret_cast<const float4*>(pB) + (size_t)b * F4_PER_BATCH;

  // Stage this batch (32 KB) into LDS via CDNA5 async loads. Each thread both
  // loads and consumes chunks i*TPB+t, so a per-wave ASYNCcnt wait suffices.
#pragma unroll
  for (int i = 0; i < CHUNKS; ++i) {
    const int idx = i * TPB + t;
    async_load_f4((unsigned int)(size_t)&sA[idx],
                  (unsigned long long)(size_t)&gA[idx]);
    async_load_f4((unsigned int)(size_t)&sB[idx],
                  (unsigned long long)(size_t)&gB[idx]);
  }
  wait_async_zero();

  // 8 streaming accumulators: raw sums + raw cross-products.
  float sax = 0.f, say = 0.f, sbx = 0.f, sby = 0.f;
  float pxx = 0.f, pxy = 0.f, pyx = 0.f, pyy = 0.f;
#pragma unroll
  for (int i = 0; i < CHUNKS; ++i) {
    const int idx = i * TPB + t;
    const float4 a  = sA[idx];   // {x0,y0,x1,y1}
    const float4 bb = sB[idx];
    sax += a.x + a.z;   say += a.y + a.w;
    sbx += bb.x + bb.z; sby += bb.y + bb.w;
    pxx += a.x * bb.x + a.z * bb.z;
    pxy += a.x * bb.y + a.z * bb.w;
    pyx += a.y * bb.x + a.w * bb.z;
    pyy += a.y * bb.y + a.w * bb.w;
  }

  // Exact f32 wave reduction on the matrix pipe (EXEC is all-1s here:
  // no divergence has occurred in this kernel yet).
  sax = wave_reduce_wmma(sax);
  say = wave_reduce_wmma(say);
  sbx = wave_reduce_wmma(sbx);
  sby = wave_reduce_wmma(sby);
  pxx = wave_reduce_wmma(pxx);
  pxy = wave_reduce_wmma(pxy);
  pyx = wave_reduce_wmma(pyx);
  pyy = wave_reduce_wmma(pyy);

  const int wave = t >> 5;
  if ((t & 31) == 0) {
    red[wave][0] = sax; red[wave][1] = say;
    red[wave][2] = sbx; red[wave][3] = sby;
    red[wave][4] = pxx; red[wave][5] = pxy;
    red[wave][6] = pyx; red[wave][7] = pyy;
  }
  __syncthreads();

  if (t == 0) {
    float acc[8];
#pragma unroll
    for (int a2 = 0; a2 < 8; ++a2) {
      float s = 0.f;
#pragma unroll
      for (int w = 0; w < 8; ++w) s += red[w][a2];
      acc[a2] = s;
    }
    const float inv = 1.0f / (float)NPTS;
    const float GAx = acc[0] * inv, GAy = acc[1] * inv;
    const float GBx = acc[2] * inv, GBy = acc[3] * inv;
    // Centered cross-covariance: H = P - Sa*Sb^T/N
    const float H00 = acc[4] - acc[0] * acc[2] * inv;
    const float H01 = acc[5] - acc[0] * acc[3] * inv;
    const float H10 = acc[6] - acc[1] * acc[2] * inv;
    const float H11 = acc[7] - acc[1] * acc[3] * inv;
    // Closed-form 2x2 Kabsch: R = rot(theta), cos ~ H00+H11, sin ~ H01-H10.
    const float cn = H00 + H11;
    const float sn = H01 - H10;
    const float r  = sqrtf(cn * cn + sn * sn);
    const float c  = cn / r;
    const float s  = sn / r;
    const float cc = fminf(fmaxf(c, -1.0f), 1.0f);
    const float theta = (s >= 0.0f ? 1.0f : -1.0f) * acosf(cc);  // == acos(tr/2)*sign(R10)
    const float tx = GBx - (c * GAx - s * GAy);
    const float ty = GBy - (s * GAx + c * GAy);
    out[b]              = theta;  // theta
    out[nbatch + b]     = tx;     // t[:,0]
    out[2 * nbatch + b] = ty;     // t[:,1]
  }
}

extern "C" void kernel_launch(void* const* d_in, const int* in_sizes, int n_in,
                              void* d_out, int out_size, void* d_ws, size_t ws_size,
                              hipStream_t stream) {
  (void)n_in; (void)out_size; (void)d_ws; (void)ws_size;
  const float* A = (const float*)d_in[0];   // pstA [B, NUM, 2] f32
  const float* B = (const float*)d_in[1];   // pstB [B, NUM, 2] f32
  float* out = (float*)d_out;               // [theta(B) | tx(B) | ty(B)]
  const int nbatch = in_sizes[0] / (NPTS * 2);
  kabsch2d_kernel<<<nbatch, TPB, 0, stream>>>(A, B, out, nbatch);
}